// FragmentPosDecoder_83786222011216
// MI455X (gfx1250) — compile-verified
//
#include <hip/hip_runtime.h>
#include <hip/hip_bf16.h>

#define NC     30000
#define NQ     1024
#define F0C    256
#define F1C    64
#define ECH    64
#define KNNK   32
#define MAXF   8
#define NCOMP  8
#define CUT    10.0f
#define EPSF   1e-6f
#define PIF    3.14159265358979323846f

typedef float v2f __attribute__((ext_vector_type(2)));
typedef float v8f __attribute__((ext_vector_type(8)));
typedef const float __attribute__((address_space(1)))* gfp;   // explicit global ptr -> global_load
typedef const int   __attribute__((address_space(1)))* gip;

__device__ __forceinline__ gfp gg(const float* p) { return (gfp)(unsigned long long)p; }
__device__ __forceinline__ gip gi(const int*   p) { return (gip)(unsigned long long)p; }

struct GVL { const float *Wg, *Ws, *Wv1, *Wv2, *bg; };
struct GVP { const float *Wdir; GVL gv; };
struct SingleAtom { GVL gv2; GVP gvp1; GVL hasatom, logsigma, mu, pi; };
struct Msg { const float *We2n,*Wn2e,*Wsca,*Wvn,*be2n,*bn2e,*bsca; GVP edge_gvp; GVL node_gv, out_gv; };

// ---------------- WMMA f32 16x16x4 tile GEMM (LDS via offsets, weights via global ptrs) ----
__device__ __forceinline__ v8f wmma4(v2f a, v2f b, v8f c) {
  return __builtin_amdgcn_wmma_f32_16x16x4_f32(false, a, false, b, (short)0, c, false, false);
}

// A: LDS row-major [M][lda] at element offset aO; fragment rows m0..m0+15, k-chunk k0..k0+3
__device__ __forceinline__ v2f fragA(int aO, int lda, int m0, int k0, int lane) {
  extern __shared__ float L[];
  int m  = m0 + (lane & 15);
  int kb = k0 + ((lane >> 4) << 1);
  int ix = aO + m * lda + kb;
  v2f a; a.x = L[ix]; a.y = L[ix + 1]; return a;
}

// W: global row-major [N][K]; we need B = W^T, cols n0..n0+15 (guard n < Ntot)
__device__ __forceinline__ v2f fragB(gfp W, int ldw, int Ntot, int n0, int k0, int lane) {
  int n  = n0 + (lane & 15);
  int kb = k0 + ((lane >> 4) << 1);
  v2f b;
  if (n < Ntot) { b.x = W[n * ldw + kb]; b.y = W[n * ldw + kb + 1]; }
  else          { b.x = 0.f;             b.y = 0.f; }
  return b;
}

// C[(MT*16) x N] = A @ W^T ; B fragment loaded once per k-step, reused across MT row-tiles.
template<int MT>
__device__ void gemmT(int aO, int lda, gfp W, int K, int cO, int ldc, int N, int lane) {
  extern __shared__ float L[];
  for (int n0 = 0; n0 < N; n0 += 16) {
    v8f acc[MT];
#pragma unroll
    for (int t = 0; t < MT; ++t) {
      v8f z = {0.f,0.f,0.f,0.f,0.f,0.f,0.f,0.f};
      acc[t] = z;
    }
    for (int k0 = 0; k0 < K; k0 += 4) {
      v2f b = fragB(W, K, N, n0, k0, lane);
#pragma unroll
      for (int t = 0; t < MT; ++t) {
        v2f a = fragA(aO, lda, t * 16, k0, lane);
        acc[t] = wmma4(a, b, acc[t]);
      }
    }
    int n = n0 + (lane & 15);
    if (n < N) {
      int mb = (lane >> 4) << 3;
#pragma unroll
      for (int t = 0; t < MT; ++t)
#pragma unroll
        for (int r = 0; r < 8; ++r)
          L[cO + (t * 16 + mb + r) * ldc + n] = acc[t][r];
    }
  }
}

// ---------------- GVP building blocks (16-row tiles, hid == inv == 64) ----------------
// vin: LDS "rowmajor48" [48][64] (row = i*3+d); sin: [16][ins]; sout: [16][outs]; vout: [48][outv]
__device__ void gv_linear_tile(GVL p, int ins, int outs, int outv, bool want_v,
                               int sinO, int vinO, int soutO, int voutO,
                               int vhO, int catO, int gateO, int lane) {
  extern __shared__ float L[];
  gemmT<3>(vinO, 64, gg(p.Wv1), 64, vhO, 64, 64, lane);        // v = Wv1 * vec
  __syncthreads();
  int cw = 64 + ins;
  for (int i = lane; i < 16*64; i += 32) {                     // vnorm -> cat[:,0:64]
    int r = i >> 6, h = i & 63;
    float a = L[vhO+(r*3+0)*64+h], b = L[vhO+(r*3+1)*64+h], c = L[vhO+(r*3+2)*64+h];
    L[catO + r*cw + h] = sqrtf(a*a + b*b + c*c);
  }
  for (int i = lane; i < 16*ins; i += 32) {                    // sca -> cat[:,64:]
    int r = i / ins, c = i - r*ins;
    L[catO + r*cw + 64 + c] = L[sinO + r*ins + c];
  }
  __syncthreads();
  gemmT<1>(catO, cw, gg(p.Ws), cw, soutO, outs, outs, lane);   // out_s
  __syncthreads();
  if (!want_v) return;
  gemmT<1>(soutO, outs, gg(p.Wg), outs, gateO, outv, outv, lane);
  __syncthreads();
  gfp bg = gg(p.bg);
  for (int i = lane; i < 16*outv; i += 32) {
    int n = i % outv;
    L[gateO + i] = 1.f / (1.f + expf(-(L[gateO + i] + bg[n])));
  }
  __syncthreads();
  gemmT<3>(vhO, 64, gg(p.Wv2), 64, voutO, outv, outv, lane);   // out_v
  __syncthreads();
  for (int i = lane; i < 48*outv; i += 32) {
    int r = i / outv, o = i - r*outv;
    L[voutO + i] = L[voutO + i] * L[gateO + (r/3)*outv + o];
  }
  __syncthreads();
}

__device__ void leaky_tile(int sO, int n, int lane) {
  extern __shared__ float L[];
  for (int i = lane; i < n; i += 32) { float x = L[sO+i]; L[sO+i] = (x >= 0.f) ? x : 0.01f*x; }
  __syncthreads();
}

__device__ void vn_leaky_tile(const float* Wdir, int vO, int dO, int lane) {
  extern __shared__ float L[];
  gemmT<3>(vO, 64, gg(Wdir), 64, dO, 64, 64, lane);
  __syncthreads();
  for (int i = lane; i < 16*64; i += 32) {
    int r = i >> 6, o = i & 63;
    float x0=L[vO+(r*3+0)*64+o], x1=L[vO+(r*3+1)*64+o], x2=L[vO+(r*3+2)*64+o];
    float d0=L[dO+(r*3+0)*64+o], d1=L[dO+(r*3+1)*64+o], d2=L[dO+(r*3+2)*64+o];
    float dot = x0*d0 + x1*d1 + x2*d2;
    float o0, o1, o2;
    if (dot >= 0.f) { o0 = x0; o1 = x1; o2 = x2; }
    else {
      float t = dot / (d0*d0 + d1*d1 + d2*d2 + EPSF);
      o0 = 0.2f*x0 + 0.8f*(x0 - t*d0);
      o1 = 0.2f*x1 + 0.8f*(x1 - t*d1);
      o2 = 0.2f*x2 + 0.8f*(x2 - t*d2);
    }
    L[vO+(r*3+0)*64+o]=o0; L[vO+(r*3+1)*64+o]=o1; L[vO+(r*3+2)*64+o]=o2;
  }
  __syncthreads();
}

__device__ void edge_gvp_tile(GVP p, int sinO, int vinO, int esO, int evO,
                              int vhO, int catO, int gateO, int lane) {
  gv_linear_tile(p.gv, 64, 64, 64, true, sinO, vinO, esO, evO, vhO, catO, gateO, lane);
  leaky_tile(esO, 16*64, lane);
  vn_leaky_tile(p.Wdir, evO, vhO, lane);
}

// message middle+tail: ES/EV edge feats, NS[16x256]->ys, NV[48x64]->yv, writes SB/VB
__device__ void msg_mid(Msg mp, int esO, int evO, int nsO, int nvO,
                        int coefO, int sbO, int vbO,
                        int vhO, int catO, int gateO, int lane) {
  extern __shared__ float L[];
  gemmT<1>(esO, 64,  gg(mp.We2n), 64,  coefO,        64,  64,  lane);   // coef1
  gemmT<1>(nsO, 256, gg(mp.Wn2e), 256, coefO + 1024, 64,  64,  lane);   // coef2
  gemmT<1>(esO, 64,  gg(mp.Wsca), 64,  sbO,          256, 256, lane);   // es @ Wsca^T
  __syncthreads();
  gfp be2n = gg(mp.be2n), bn2e = gg(mp.bn2e), bsca = gg(mp.bsca);
  for (int i = lane; i < 16*64; i += 32) {
    int o = i & 63;
    L[coefO + i]        += be2n[o];
    L[coefO + 1024 + i] += bn2e[o];
  }
  for (int i = lane; i < 16*256; i += 32) {
    int c = i & 255;
    L[nsO + i] = L[nsO + i] * (L[sbO + i] + bsca[c]);                    // y_s
  }
  __syncthreads();
  gemmT<3>(evO, 64, gg(mp.Wvn), 64, vhO, 64, 64, lane);                  // Wvn * ev
  __syncthreads();
  for (int i = lane; i < 48*64; i += 32) {
    int r = i >> 6, o = i & 63, e = r / 3;
    L[nvO + i] = L[coefO + e*64 + o] * L[nvO + i]
               + L[coefO + 1024 + e*64 + o] * L[vhO + i];                // y_v
  }
  __syncthreads();
  gv_linear_tile(mp.out_gv, 256, 256, 64, true, nsO, nvO, sbO, vbO, vhO, catO, gateO, lane);
}

__device__ __forceinline__ float cutoffC(float d) {
  return (d <= CUT) ? 0.5f * (cosf(d * PIF / CUT) + 1.f) : 0.f;
}

// ---------------- kernels ----------------
__global__ void k_zero(float* a, int n) {
  int i = blockIdx.x * 256 + threadIdx.x; if (i < n) a[i] = 0.f;
}

__global__ void k_init(const float* h_sca, const float* h_vec, const float* pos_compose,
                       const int* index_query, float* hf_s, float* hf_v, float* posf) {
  int i = blockIdx.x * 256 + threadIdx.x;
  const int per = F0C + 192 + 3;
  if (i >= NQ * per) return;
  int q = i / per, r = i - q * per;
  int idx = gi(index_query)[q];
  if (r < F0C)            hf_s[q*F0C + r] = gg(h_sca)[idx*F0C + r];
  else if (r < F0C + 192) { int rr = r - F0C; hf_v[q*192 + rr] = gg(h_vec)[idx*192 + rr]; }
  else                    { int d = r - F0C - 192; posf[q*3 + d] = gg(pos_compose)[idx*3 + d]; }
}

__global__ void k_blend0(const float* hg_s, const float* hg_v, const float* pos_gen_true,
                         const unsigned char* mask, float* hf_s, float* hf_v, float* posf) {
  int i = blockIdx.x * 256 + threadIdx.x;
  const int per = F0C + 192 + 3;
  if (i >= NQ * per) return;
  int q = i / per, r = i - q * per;
  bool cont = mask[q*MAXF + 1] != 0;
  if (r < F0C)            { if (cont) hf_s[q*F0C + r] = hg_s[q*F0C + r]; }
  else if (r < F0C + 192) { int rr = r - F0C; if (cont) hf_v[q*192 + rr] = hg_v[q*192 + rr]; }
  else                    { int d = r - F0C - 192; posf[q*3 + d] = pos_gen_true[(q*MAXF + 0)*3 + d]; }
}

__global__ void k_knn(int step, const float* pos_compose_, const float* pos_gen_true,
                      const int* batch_compose_, const int* batch_query, int* nbr) {
  int q = blockIdx.x, t = threadIdx.x;
  __shared__ float red_d[256];
  __shared__ int   red_i[256];
  __shared__ float s_prevd;
  __shared__ int   s_previ;
  gfp pc = gg(pos_compose_);
  gip bc = gi(batch_compose_);
  float qx = pos_gen_true[(q*MAXF+step)*3+0];
  float qy = pos_gen_true[(q*MAXF+step)*3+1];
  float qz = pos_gen_true[(q*MAXF+step)*3+2];
  int bq = batch_query[q];
  if (t == 0) { s_prevd = -1.f; s_previ = -1; }
  __syncthreads();
  for (int r = 0; r < KNNK; ++r) {
    float prevd = s_prevd; int previ = s_previ;
    float bd = 3.4e38f; int bi = 0x7fffffff;
    for (int j = t; j < NC; j += 256) {
      float dx = qx - pc[j*3+0];
      float dy = qy - pc[j*3+1];
      float dz = qz - pc[j*3+2];
      float d2 = dx*dx + dy*dy + dz*dz;
      if (bc[j] != bq) d2 = 3.0e38f;
      bool excl = (d2 < prevd) || (d2 == prevd && j <= previ);
      if (!excl && (d2 < bd || (d2 == bd && j < bi))) { bd = d2; bi = j; }
    }
    red_d[t] = bd; red_i[t] = bi;
    __syncthreads();
    for (int s = 128; s > 0; s >>= 1) {
      if (t < s) {
        float od = red_d[t+s]; int oi = red_i[t+s];
        if (od < red_d[t] || (od == red_d[t] && oi < red_i[t])) { red_d[t] = od; red_i[t] = oi; }
      }
      __syncthreads();
    }
    if (t == 0) { nbr[q*KNNK + r] = red_i[0]; s_prevd = red_d[0]; s_previ = red_i[0]; }
    __syncthreads();
  }
}

// precompute message.node_gv over all compose nodes (step-invariant)
__global__ void __launch_bounds__(32) k_nodegv(const float* h_sca, const float* h_vec,
                                               GVL node_gv, float* ns, float* nv) {
  extern __shared__ float L[];
  const int lane = threadIdx.x;
  const int n0 = blockIdx.x * 16;
  const int VIN=0, SIN=3072, VH=7168, CAT=10240, SOUT=15360, VOUT=19456, GATE=22528;
  gfp hs = gg(h_sca), hv = gg(h_vec);
  for (int i = lane; i < 16*F0C; i += 32) L[SIN + i] = hs[n0*F0C + i];
  for (int i = lane; i < 16*192; i += 32) {
    int e = i/192, rem = i - e*192, c = rem/3, d = rem - c*3;
    L[VIN + (e*3+d)*64 + c] = hv[((n0+e)*64 + c)*3 + d];
  }
  __syncthreads();
  gv_linear_tile(node_gv, 256, 256, 64, true, SIN, VIN, SOUT, VOUT, VH, CAT, GATE, lane);
  for (int i = lane; i < 16*F0C; i += 32) ns[n0*F0C + i] = L[SOUT + i];
  for (int i = lane; i < 16*192; i += 32) {
    int e = i/192, rem = i - e*192, o = rem/3, d = rem - o*3;
    nv[((n0+e)*64 + o)*3 + d] = L[VOUT + (e*3+d)*64 + o];
  }
}

// fused gather + message for one 16-edge half-tile of one query
__global__ void __launch_bounds__(32) k_msg(int step, const int* nbr_, const float* pos_compose_,
      const float* pos_gen_true, const float* wexp_,
      const float* ns_g_, const float* nv_g_, Msg mp, float* hg_s, float* hg_v) {
  extern __shared__ float L[];
  const int lane = threadIdx.x;
  const int tile = blockIdx.x, q = tile >> 1, e0 = (tile & 1) * 16;
  const int VINE=0, SINE=3072, VH=4096, CAT=7168, GATE=12288, ES=13312, EV=14336,
            NS=17408, NV=21504, COEF=24576, SB=26624, VB=30720, DST=33792;
  gip nbr = gi(nbr_);
  gfp pc = gg(pos_compose_), wexp = gg(wexp_), ns_g = gg(ns_g_), nv_g = gg(nv_g_);
  float gx = pos_gen_true[(q*MAXF+step)*3+0];
  float gy = pos_gen_true[(q*MAXF+step)*3+1];
  float gz = pos_gen_true[(q*MAXF+step)*3+2];
  if (lane < 16) {
    int j = nbr[q*KNNK + e0 + lane];
    float dx = gx - pc[j*3+0];
    float dy = gy - pc[j*3+1];
    float dz = gz - pc[j*3+2];
    float dn = sqrtf(dx*dx + dy*dy + dz*dz);
    L[DST + lane] = dn;
    float inv = 1.f / (dn + 1e-7f);
    L[COEF + lane*3+0] = dx*inv; L[COEF + lane*3+1] = dy*inv; L[COEF + lane*3+2] = dz*inv;
  }
  __syncthreads();
  const float so = CUT / 63.f;
  const float gc = -0.5f / (so * so);
  for (int i = lane; i < 16*ECH; i += 32) {
    int e = i >> 6, c = i & 63;
    float diff = L[DST + e] - so * (float)c;
    L[SINE + i] = expf(gc * diff * diff);
  }
  for (int i = lane; i < 16*192; i += 32) {
    int e = i/192, rem = i - e*192, c = rem/3, d = rem - c*3;
    L[VINE + (e*3+d)*64 + c] = wexp[c] * L[COEF + e*3+d];
  }
  __syncthreads();
  edge_gvp_tile(mp.edge_gvp, SINE, VINE, ES, EV, VH, CAT, GATE, lane);
  for (int i = lane; i < 16*F0C; i += 32) {
    int e = i >> 8, c = i & 255;
    L[NS + i] = ns_g[nbr[q*KNNK + e0 + e]*F0C + c];
  }
  for (int i = lane; i < 16*192; i += 32) {
    int e = i/192, rem = i - e*192, o = rem/3, d = rem - o*3;
    L[NV + (e*3+d)*64 + o] = nv_g[(nbr[q*KNNK + e0 + e]*64 + o)*3 + d];
  }
  __syncthreads();
  msg_mid(mp, ES, EV, NS, NV, COEF, SB, VB, VH, CAT, GATE, lane);
  for (int n = lane; n < F0C; n += 32) {
    float acc = 0.f;
    for (int e = 0; e < 16; ++e) acc += cutoffC(L[DST + e]) * L[SB + e*F0C + n];
    atomicAdd(&hg_s[q*F0C + n], acc);
  }
  for (int i = lane; i < 192; i += 32) {
    int o = i/3, d = i - o*3;
    float acc = 0.f;
    for (int e = 0; e < 16; ++e) acc += cutoffC(L[DST + e]) * L[VB + (e*3+d)*64 + o];
    atomicAdd(&hg_v[(q*64 + o)*3 + d], acc);
  }
}

// update_focal (steps >= 1): node_gv on hg + message + masked accumulate into hf
__global__ void __launch_bounds__(32) k_upd(int step, const float* pos_gen_true,
      const float* wexp_, Msg mp, const float* hg_s_g, const float* hg_v_g,
      const float* posf, float* hf_s, float* hf_v, const unsigned char* mask) {
  extern __shared__ float L[];
  const int lane = threadIdx.x;
  const int q0 = blockIdx.x * 16;
  const int SIN2=0, VIN2=4096, VH=7168, CAT=10240, GATE=15360, NS=16384, NV=20480,
            VINE=23552, SINE=26624, ES=27648, EV=28672, COEF=31744, SB=33792, VB=37888,
            DST=40960, UV=40976;
  gfp hgs = gg(hg_s_g), hgv = gg(hg_v_g), wexp = gg(wexp_);
  for (int i = lane; i < 16*F0C; i += 32) L[SIN2 + i] = hgs[q0*F0C + i];
  for (int i = lane; i < 16*192; i += 32) {
    int e = i/192, rem = i - e*192, o = rem/3, d = rem - o*3;
    L[VIN2 + (e*3+d)*64 + o] = hgv[((q0+e)*64 + o)*3 + d];
  }
  __syncthreads();
  gv_linear_tile(mp.node_gv, 256, 256, 64, true, SIN2, VIN2, NS, NV, VH, CAT, GATE, lane);
  if (lane < 16) {
    int q = q0 + lane;
    float dx = posf[q*3+0] - pos_gen_true[(q*MAXF+step)*3+0];
    float dy = posf[q*3+1] - pos_gen_true[(q*MAXF+step)*3+1];
    float dz = posf[q*3+2] - pos_gen_true[(q*MAXF+step)*3+2];
    float dn = sqrtf(dx*dx + dy*dy + dz*dz);
    L[DST + lane] = dn;
    float inv = 1.f / (dn + 1e-7f);
    L[UV + lane*3+0] = dx*inv; L[UV + lane*3+1] = dy*inv; L[UV + lane*3+2] = dz*inv;
  }
  __syncthreads();
  const float so = CUT / 63.f;
  const float gc = -0.5f / (so * so);
  for (int i = lane; i < 16*ECH; i += 32) {
    int e = i >> 6, c = i & 63;
    float diff = L[DST + e] - so * (float)c;
    L[SINE + i] = expf(gc * diff * diff);
  }
  for (int i = lane; i < 16*192; i += 32) {
    int e = i/192, rem = i - e*192, c = rem/3, d = rem - c*3;
    L[VINE + (e*3+d)*64 + c] = wexp[c] * L[UV + e*3+d];
  }
  __syncthreads();
  edge_gvp_tile(mp.edge_gvp, SINE, VINE, ES, EV, VH, CAT, GATE, lane);
  msg_mid(mp, ES, EV, NS, NV, COEF, SB, VB, VH, CAT, GATE, lane);
  for (int i = lane; i < 16*F0C; i += 32) {
    int e = i >> 8, n = i & 255;
    int q = q0 + e;
    if (!mask[q*MAXF + step + 1]) continue;
    hf_s[q*F0C + n] += cutoffC(L[DST + e]) * L[SB + e*F0C + n];
  }
  for (int i = lane; i < 16*192; i += 32) {
    int e = i/192, rem = i - e*192, o = rem/3, d = rem - o*3;
    int q = q0 + e;
    if (!mask[q*MAXF + step + 1]) continue;
    hf_v[(q*64 + o)*3 + d] += cutoffC(L[DST + e]) * L[VB + (e*3+d)*64 + o];
  }
}

// decoder: gvp1 -> gv2 -> {mu, logsigma, pi, hasatom}
__global__ void __launch_bounds__(32) k_dec(int step, SingleAtom p,
      const float* hf_s, const float* hf_v, const float* posf,
      const unsigned char* mask, float* mus, float* sigmas, float* pis, float* has) {
  extern __shared__ float L[];
  const int lane = threadIdx.x;
  const int q0 = blockIdx.x * 16;
  const int SIN=0, VIN=4096, VH=7168, CAT=10240, GATE=15360, S1=16384, V1=20480,
            S2=24576, V2=28672, HS=31744, HV=32000;
  gfp hfs = gg(hf_s), hfv = gg(hf_v);
  for (int i = lane; i < 16*F0C; i += 32) L[SIN + i] = hfs[q0*F0C + i];
  for (int i = lane; i < 16*192; i += 32) {
    int e = i/192, rem = i - e*192, o = rem/3, d = rem - o*3;
    L[VIN + (e*3+d)*64 + o] = hfv[((q0+e)*64 + o)*3 + d];
  }
  __syncthreads();
  gv_linear_tile(p.gvp1.gv, 256, 256, 64, true, SIN, VIN, S1, V1, VH, CAT, GATE, lane);
  leaky_tile(S1, 16*256, lane);
  vn_leaky_tile(p.gvp1.Wdir, V1, VH, lane);
  gv_linear_tile(p.gv2, 256, 256, 64, true, S1, V1, S2, V2, VH, CAT, GATE, lane);

  // mu
  gv_linear_tile(p.mu, 256, 8, 8, true, S2, V2, HS, HV, VH, CAT, GATE, lane);
  for (int i = lane; i < 16*NCOMP*3; i += 32) {
    int e = i/24, rem = i - e*24, k = rem/3, d = rem - k*3;
    int q = q0 + e;
    float vf = mask[q*MAXF + step] ? 1.f : 0.f;
    mus[((q*MAXF + step)*NCOMP + k)*3 + d] = vf * (L[HV + (e*3+d)*8 + k] + posf[q*3 + d]);
  }
  __syncthreads();
  // logsigma -> sigma = exp
  gv_linear_tile(p.logsigma, 256, 8, 8, true, S2, V2, HS, HV, VH, CAT, GATE, lane);
  for (int i = lane; i < 16*NCOMP*3; i += 32) {
    int e = i/24, rem = i - e*24, k = rem/3, d = rem - k*3;
    int q = q0 + e;
    float vf = mask[q*MAXF + step] ? 1.f : 0.f;
    sigmas[((q*MAXF + step)*NCOMP + k)*3 + d] = vf * expf(L[HV + (e*3+d)*8 + k]);
  }
  __syncthreads();
  // pi (scalar head, softmax over 8)
  gv_linear_tile(p.pi, 256, 8, 1, false, S2, V2, HS, HV, VH, CAT, GATE, lane);
  if (lane < 16) {
    int q = q0 + lane;
    float vf = mask[q*MAXF + step] ? 1.f : 0.f;
    float m = -3.0e38f;
    for (int k = 0; k < 8; ++k) m = fmaxf(m, L[HS + lane*8 + k]);
    float ex[8]; float s = 0.f;
    for (int k = 0; k < 8; ++k) { ex[k] = expf(L[HS + lane*8 + k] - m); s += ex[k]; }
    for (int k = 0; k < 8; ++k) pis[(q*MAXF + step)*NCOMP + k] = vf * ex[k] / s;
  }
  __syncthreads();
  // hasatom
  gv_linear_tile(p.hasatom, 256, 1, 1, false, S2, V2, HS, HV, VH, CAT, GATE, lane);
  if (lane < 16) {
    int q = q0 + lane;
    float vf = mask[q*MAXF + step] ? 1.f : 0.f;
    has[q*MAXF + step] = vf * L[HS + lane];
  }
}

// ---------------- host ----------------
static GVL getGVL(void* const* p, int& i) {
  GVL g; g.Wg=(const float*)p[i++]; g.Ws=(const float*)p[i++]; g.Wv1=(const float*)p[i++];
  g.Wv2=(const float*)p[i++]; g.bg=(const float*)p[i++]; return g;
}
static GVP getGVP(void* const* p, int& i) { GVP g; g.Wdir=(const float*)p[i++]; g.gv=getGVL(p,i); return g; }
static SingleAtom getSA(void* const* p, int& i) {
  SingleAtom s; s.gv2=getGVL(p,i); s.gvp1=getGVP(p,i); s.hasatom=getGVL(p,i);
  s.logsigma=getGVL(p,i); s.mu=getGVL(p,i); s.pi=getGVL(p,i); return s;
}
static Msg getMsg(void* const* p, int& i) {
  Msg m; m.We2n=(const float*)p[i++]; m.Wn2e=(const float*)p[i++]; m.Wsca=(const float*)p[i++];
  m.Wvn=(const float*)p[i++]; m.be2n=(const float*)p[i++]; m.bn2e=(const float*)p[i++];
  m.bsca=(const float*)p[i++]; m.edge_gvp=getGVP(p,i); m.node_gv=getGVL(p,i); m.out_gv=getGVL(p,i);
  return m;
}

extern "C" void kernel_launch(void* const* d_in, const int* in_sizes, int n_in,
                              void* d_out, int out_size, void* d_ws, size_t ws_size,
                              hipStream_t stream) {
  (void)in_sizes; (void)n_in; (void)out_size; (void)ws_size;
  const float* h_sca        = (const float*)d_in[0];
  const float* h_vec        = (const float*)d_in[1];
  const float* pos_compose  = (const float*)d_in[2];
  const float* pos_gen_true = (const float*)d_in[3];
  const int*   index_query  = (const int*)d_in[4];
  const unsigned char* mask = (const unsigned char*)d_in[5];
  const int*   batch_compose= (const int*)d_in[6];
  const int*   batch_query  = (const int*)d_in[7];
  int i = 8;
  SingleAtom dec  = getSA(d_in, i);
  SingleAtom decp = getSA(d_in, i);
  Msg msg = getMsg(d_in, i);
  Msg upd = getMsg(d_in, i);
  const float* wexp = (const float*)d_in[i++];

  float* ws   = (float*)d_ws;
  float* ns   = ws; ws += (size_t)NC * F0C;
  float* nv   = ws; ws += (size_t)NC * 192;
  float* hf_s = ws; ws += (size_t)NQ * F0C;
  float* hf_v = ws; ws += (size_t)NQ * 192;
  float* posf = ws; ws += (size_t)NQ * 3;
  float* hg_s = ws; ws += (size_t)NQ * F0C;
  float* hg_v = ws; ws += (size_t)NQ * 192;
  int*   nbr  = (int*)ws;

  float* mus    = (float*)d_out;
  float* sigmas = mus    + (size_t)NQ * MAXF * NCOMP * 3;
  float* pis    = sigmas + (size_t)NQ * MAXF * NCOMP * 3;
  float* has    = pis    + (size_t)NQ * MAXF * NCOMP;

  const int per = F0C + 192 + 3;
  k_init<<<(NQ*per + 255)/256, 256, 0, stream>>>(h_sca, h_vec, pos_compose, index_query, hf_s, hf_v, posf);
  k_nodegv<<<NC/16, 32, 23552*sizeof(float), stream>>>(h_sca, h_vec, msg.node_gv, ns, nv);

  for (int step = 0; step < MAXF; ++step) {
    SingleAtom cur = (step == 0) ? decp : dec;
    k_dec<<<NQ/16, 32, 32768*sizeof(float), stream>>>(step, cur, hf_s, hf_v, posf, mask,
                                                      mus, sigmas, pis, has);
    if (step == MAXF - 1) break;
    k_knn<<<NQ, 256, 0, stream>>>(step, pos_compose, pos_gen_true, batch_compose, batch_query, nbr);
    k_zero<<<(NQ*448 + 255)/256, 256, 0, stream>>>(hg_s, NQ*448);  // hg_s + hg_v contiguous
    k_msg<<<NQ*2, 32, 33824*sizeof(float), stream>>>(step, nbr, pos_compose, pos_gen_true, wexp,
                                                     ns, nv, msg, hg_s, hg_v);
    if (step == 0)
      k_blend0<<<(NQ*per + 255)/256, 256, 0, stream>>>(hg_s, hg_v, pos_gen_true, mask, hf_s, hf_v, posf);
    else
      k_upd<<<NQ/16, 32, 41056*sizeof(float), stream>>>(step, pos_gen_true, wexp, upd,
                                                        hg_s, hg_v, posf, hf_s, hf_v, mask);
  }
}